// IndexedAdd_72679436583586
// MI455X (gfx1250) — compile-verified
//
#include <hip/hip_runtime.h>
#include <hip/hip_bf16.h>

#define N_NODES 100000
#define N_EDGES 500000
#define D_FEAT  256

#ifndef __has_builtin
#define __has_builtin(x) 0
#endif

// Pointer types matching the gfx1250 async builtin signature
// (param is "int __vector(4)" pointer; AS1 global / AS3 LDS).
typedef int v4i __attribute__((ext_vector_type(4)));
typedef __attribute__((address_space(1))) v4i as1_v4i;
typedef __attribute__((address_space(3))) v4i as3_v4i;

// Non-returning f32 atomic add -> global_atomic_add_f32 at L2.
__device__ __forceinline__ void gatomic_fadd(float* p, float v) {
  unsafeAtomicAdd(p, v);  // native HW f32 atomic (no CAS loop)
}

// ---------------------------------------------------------------------------
// (Placed first in the file so its disassembly shows in the feedback window.)
// One wave32 per edge, EPW edges per wave:
//   - indices & weight broadcast to SGPRs via v_readfirstlane_b32
//   - 2x global_load_b128 per lane covers the 1KB src row, fully coalesced
//   - 8x non-returning global_atomic_add_f32 per lane into the dst row
//   - global_prefetch_b8 of the NEXT edge's src row hides gather latency
// ---------------------------------------------------------------------------
__global__ __launch_bounds__(256) void IndexedAdd_scatter_kernel(
    const float* __restrict__ src, const float* __restrict__ weight,
    const int* __restrict__ index, float* out) {
  constexpr int LANES = 32;   // gfx1250 is wave32
  constexpr int EPW   = 4;    // edges per wave
  const int lane = threadIdx.x & (LANES - 1);
  const int wave = threadIdx.x >> 5;
  const int wavesPerBlock = blockDim.x >> 5;
  const int e0 = (blockIdx.x * wavesPerBlock + wave) * EPW;

  const int* idx_src = index;            // row 0: gather indices
  const int* idx_dst = index + N_EDGES;  // row 1: scatter indices

#pragma unroll
  for (int k = 0; k < EPW; ++k) {
    const int e = e0 + k;
    if (e >= N_EDGES) return;

    // Prefetch next edge's source row (covers the 1KB row's cachelines).
    if (k + 1 < EPW && e + 1 < N_EDGES) {
      const int sn = __builtin_amdgcn_readfirstlane(idx_src[e + 1]);
      __builtin_prefetch(src + (size_t)sn * D_FEAT + lane * 8, 0, 1);
    }

    const int s = __builtin_amdgcn_readfirstlane(idx_src[e]);
    const int d = __builtin_amdgcn_readfirstlane(idx_dst[e]);
    const float w = __uint_as_float(
        (unsigned)__builtin_amdgcn_readfirstlane((int)__float_as_uint(weight[e])));

    const float4* srow = (const float4*)(src + (size_t)s * D_FEAT);
    float4 a = srow[lane];          // floats [lane*4     .. lane*4+3]
    float4 b = srow[lane + LANES];  // floats [128+lane*4 .. 128+lane*4+3]

    float* orow = out + (size_t)d * D_FEAT + lane * 4;
    gatomic_fadd(orow + 0, a.x * w);
    gatomic_fadd(orow + 1, a.y * w);
    gatomic_fadd(orow + 2, a.z * w);
    gatomic_fadd(orow + 3, a.w * w);
    gatomic_fadd(orow + 128 + 0, b.x * w);
    gatomic_fadd(orow + 128 + 1, b.y * w);
    gatomic_fadd(orow + 128 + 2, b.z * w);
    gatomic_fadd(orow + 128 + 3, b.w * w);
  }
}

// ---------------------------------------------------------------------------
// out = dst  (deterministic re-init every call; harness does not re-poison).
// Uses the CDNA5 async global<->LDS DMA path (ASYNCcnt):
// global_load_async_to_lds_b128 / global_store_async_from_lds_b128,
// streaming 16B/lane without touching VGPRs. Grid sized so each thread
// executes exactly one iteration (no per-wave wait serialization).
// ---------------------------------------------------------------------------
__global__ __launch_bounds__(256) void IndexedAdd_copy_kernel(
    const float* __restrict__ in, float* __restrict__ out, long n_vec4) {
#if __has_builtin(__builtin_amdgcn_global_load_async_to_lds_b128) && \
    __has_builtin(__builtin_amdgcn_global_store_async_from_lds_b128) && \
    __has_builtin(__builtin_amdgcn_s_wait_asynccnt)
  __shared__ float4 stage[256];
  const int tid = threadIdx.x;
  as3_v4i* lds_slot = (as3_v4i*)&stage[tid];
  size_t i = (size_t)blockIdx.x * blockDim.x + tid;
  const size_t stride = (size_t)gridDim.x * blockDim.x;
  for (; i < (size_t)n_vec4; i += stride) {
    as1_v4i* gsrc = (as1_v4i*)((const float4*)in + i);
    as1_v4i* gdst = (as1_v4i*)((float4*)out + i);
    __builtin_amdgcn_global_load_async_to_lds_b128(gsrc, lds_slot, 0, 0);
    __builtin_amdgcn_s_wait_asynccnt(0);   // LDS slot filled
    __builtin_amdgcn_global_store_async_from_lds_b128(gdst, lds_slot, 0, 0);
    __builtin_amdgcn_s_wait_asynccnt(0);   // LDS slot free for next iter
  }
#else
  size_t i = (size_t)blockIdx.x * blockDim.x + threadIdx.x;
  const size_t stride = (size_t)gridDim.x * blockDim.x;
  for (; i < (size_t)n_vec4; i += stride)
    ((float4*)out)[i] = ((const float4*)in)[i];
#endif
}

extern "C" void kernel_launch(void* const* d_in, const int* in_sizes, int n_in,
                              void* d_out, int out_size, void* d_ws, size_t ws_size,
                              hipStream_t stream) {
  const float* dst    = (const float*)d_in[0];  // [N_NODES, D_FEAT]
  const float* src    = (const float*)d_in[1];  // [N_NODES, D_FEAT]
  const float* weight = (const float*)d_in[2];  // [N_EDGES, 1]
  const int*   index  = (const int*)d_in[3];    // [2, N_EDGES]
  float* out = (float*)d_out;

  // 1) out = dst  (6.4M float4, exactly 25000 blocks x 256 threads)
  const long n_vec4 = (long)N_NODES * D_FEAT / 4;
  hipLaunchKernelGGL(IndexedAdd_copy_kernel, dim3(25000), dim3(256), 0, stream,
                     dst, out, n_vec4);

  // 2) gather * weight, scatter-add via L2 f32 atomics
  const int edgesPerBlock = (256 / 32) * 4;  // 8 waves x 4 edges = 32
  const int blocks = (N_EDGES + edgesPerBlock - 1) / edgesPerBlock;  // 15625
  hipLaunchKernelGGL(IndexedAdd_scatter_kernel, dim3(blocks), dim3(256), 0, stream,
                     src, weight, index, out);
}